// Segmenter_1984274891517
// MI455X (gfx1250) — compile-verified
//
#include <hip/hip_runtime.h>
#include <cstdint>
#include <cstddef>

// ---------------------------------------------------------------------------
// PointTransformer segmentation forward pass for MI455X (gfx1250, wave32).
// All matmuls go through V_WMMA_F32_16X16X4_F32 (full fp32 precision; the
// workload is latency/gather bound, so fp32 WMMA is the right point).
// GEMM inner loop: branch-free clamped addressing, 32-bit offsets off uniform
// SGPR bases (scale_offset addressing), software-pipelined K loop (fragments
// for step kb+4 load before the WMMAs of step kb issue).
// ---------------------------------------------------------------------------

#define N0 8192
#define N1 2048
#define N2 512
#define C0 64
#define C1 128
#define C2 256
#define KNBR 16
#define KE 17          // K neighbors + self loop
#define KUP 3
#define HIDN 64
#define NCLS 8
#define EPSN 1e-5f
#define E0 (N0 * KE)

typedef float v2f __attribute__((ext_vector_type(2)));
typedef float v8f __attribute__((ext_vector_type(8)));

static inline int ceil_div(int a, int b) { return (a + b - 1) / b; }

// ---------------------------------------------------------------------------
// Generic GEMM: C[M,N] = act(A[M,K] @ W[K,N] + bias), row-major.
// One wave per 16x(16*NT) output strip, K stepped by 4 via WMMA f32 16x16x4.
// Fragment layouts per CDNA5 ISA 7.12.2:
//   A 16x4 f32: lane(0..15) m=lane, vgpr0/1 = k=(2*half)+0/1  (half=lane>=16)
//   B 4x16 f32: lane n=lane&15,      vgpr0/1 = k=(2*half)+0/1
//   C/D:        vgpr r -> m = r + 8*half, n = lane&15
// Out-of-range m/n are address-clamped; they only touch accumulator lanes
// masked at writeback. Only the K remainder (K=2,3 layers) needs value masks.
// ---------------------------------------------------------------------------
template <int NT>
__global__ __launch_bounds__(256) void gemm_kernel(
    const float* __restrict__ A, const float* __restrict__ W,
    const float* __restrict__ bias, float* __restrict__ C,
    int M, int Kc, int Nc, int relu)
{
  int wid  = blockIdx.x * (blockDim.x >> 5) + ((int)threadIdx.x >> 5);
  int lane = (int)threadIdx.x & 31;
  int tnc  = (Nc + 16 * NT - 1) / (16 * NT);
  int tiles = ((M + 15) >> 4) * tnc;
  if (wid >= tiles) return;                 // wave-uniform: EXEC stays all-ones
  int tm   = (wid / tnc) << 4;
  int tn   = (wid % tnc) * (16 * NT);
  int half = lane >> 4;
  int l16  = lane & 15;

  // 32-bit offsets from uniform bases (all arrays < 2^31 elements).
  int m_a  = tm + l16;
  int aoff = (m_a < M ? m_a : (M - 1)) * Kc;
  int nidx[NT], ncl[NT];
#pragma unroll
  for (int t = 0; t < NT; ++t) {
    nidx[t] = tn + 16 * t + l16;
    ncl[t]  = nidx[t] < Nc ? nidx[t] : (Nc - 1);
  }

  v8f acc[NT];
#pragma unroll
  for (int t = 0; t < NT; ++t) acc[t] = (v8f){0.f,0.f,0.f,0.f,0.f,0.f,0.f,0.f};

  int Kfull = Kc & ~3;
  if (Kfull > 0) {
    if ((Kc & 1) == 0) {
      // Even K (all full-tile layers): A fragment is one 8B-aligned b64 load.
      const v2f* A2 = (const v2f*)A;
      int a2base = (aoff >> 1) + half;      // float2 index at kb=0, ka=2*half
      int hrow   = (2 * half) * Nc;
      // Preload kb = 0.
      v2f afc = A2[a2base];
      v2f bfc[NT];
#pragma unroll
      for (int t = 0; t < NT; ++t) {
        bfc[t].x = W[ncl[t] + hrow];
        bfc[t].y = W[ncl[t] + hrow + Nc];
      }
      for (int kb = 0; kb < Kfull; kb += 4) {
        // Issue next step's loads first (clamped: last iter reloads current).
        int kn  = (kb + 4 < Kfull) ? (kb + 4) : kb;
        int wrn = kn * Nc + hrow;
        v2f afn = A2[a2base + (kn >> 1)];
        v2f bfn[NT];
#pragma unroll
        for (int t = 0; t < NT; ++t) {
          bfn[t].x = W[ncl[t] + wrn];
          bfn[t].y = W[ncl[t] + wrn + Nc];
        }
        // Consume current fragments while next loads are in flight.
#pragma unroll
        for (int t = 0; t < NT; ++t)
          acc[t] = __builtin_amdgcn_wmma_f32_16x16x4_f32(
              false, afc, false, bfc[t], (short)0, acc[t], false, false);
        afc = afn;
#pragma unroll
        for (int t = 0; t < NT; ++t) bfc[t] = bfn[t];
      }
    } else {
      // Odd K >= 4 (not used by this net; kept for generality).
      for (int kb = 0; kb < Kfull; kb += 4) {
        int ka = kb + 2 * half;
        v2f af;
        af.x = A[aoff + ka];
        af.y = A[aoff + ka + 1];
        int wrow = ka * Nc;
#pragma unroll
        for (int t = 0; t < NT; ++t) {
          v2f bf;
          bf.x = W[ncl[t] + wrow];
          bf.y = W[ncl[t] + wrow + Nc];
          acc[t] = __builtin_amdgcn_wmma_f32_16x16x4_f32(
              false, af, false, bf, (short)0, acc[t], false, false);
        }
      }
    }
  }

  if (Kfull < Kc) {
    // Masked remainder step (only K=2,3 layers): clamped addresses, value
    // selects (cndmask) for k overflow — no exec-divergent loads.
    int ka0 = Kfull + 2 * half;
    int ka1 = ka0 + 1;
    int c0 = ka0 < Kc, c1 = ka1 < Kc;
    int k0 = c0 ? ka0 : (Kc - 1);
    int k1 = c1 ? ka1 : (Kc - 1);
    v2f af;
    af.x = A[aoff + k0];
    af.y = A[aoff + k1];
    af.x = c0 ? af.x : 0.f;
    af.y = c1 ? af.y : 0.f;
#pragma unroll
    for (int t = 0; t < NT; ++t) {
      v2f bf;
      bf.x = W[ncl[t] + k0 * Nc];
      bf.y = W[ncl[t] + k1 * Nc];
      bf.x = c0 ? bf.x : 0.f;
      bf.y = c1 ? bf.y : 0.f;
      acc[t] = __builtin_amdgcn_wmma_f32_16x16x4_f32(
          false, af, false, bf, (short)0, acc[t], false, false);
    }
  }

#pragma unroll
  for (int t = 0; t < NT; ++t) {
    if (nidx[t] < Nc) {
      float bv = bias ? bias[nidx[t]] : 0.f;
#pragma unroll
      for (int r = 0; r < 8; ++r) {
        int m = tm + r + (half << 3);
        if (m < M) {
          float v = acc[t][r] + bv;
          if (relu) v = fmaxf(v, 0.f);
          C[m * Nc + nidx[t]] = v;
        }
      }
    }
  }
}

// ---------------------------------------------------------------------------
// Per-column mean / rsqrt(var+eps) over M rows (batch/instance norm stats).
// ---------------------------------------------------------------------------
__global__ __launch_bounds__(256) void colstats_kernel(
    const float* __restrict__ H, int M, int C,
    float* __restrict__ mu, float* __restrict__ rinv)
{
  __shared__ float s1[256];
  __shared__ float s2[256];
  int c = blockIdx.x;
  float a = 0.f, b = 0.f;
  for (int i = (int)threadIdx.x; i < M; i += (int)blockDim.x) {
    float v = H[i * C + c];
    a += v; b += v * v;
  }
  s1[threadIdx.x] = a; s2[threadIdx.x] = b;
  __syncthreads();
  for (int s = (int)blockDim.x >> 1; s > 0; s >>= 1) {
    if ((int)threadIdx.x < s) {
      s1[threadIdx.x] += s1[threadIdx.x + s];
      s2[threadIdx.x] += s2[threadIdx.x + s];
    }
    __syncthreads();
  }
  if (threadIdx.x == 0) {
    float m   = s1[0] / (float)M;
    float var = s2[0] / (float)M - m * m;
    mu[c]   = m;
    rinv[c] = rsqrtf(fmaxf(var, 0.f) + EPSN);
  }
}

// Apply norm (+optional affine g/b, +optional ReLU) in place.
__global__ __launch_bounds__(256) void bnact_kernel(
    float* __restrict__ H, int total, int C,
    const float* __restrict__ mu, const float* __restrict__ rinv,
    const float* __restrict__ g, const float* __restrict__ b, int relu)
{
  int t = blockIdx.x * blockDim.x + threadIdx.x;
  if (t >= total) return;
  int c = t % C;
  float v = (H[t] - mu[c]) * rinv[c];
  if (g) v = v * g[c] + b[c];
  if (relu) v = fmaxf(v, 0.f);
  H[t] = v;
}

// ---------------------------------------------------------------------------
// Brute-force kNN in 2D: per query thread keeps a KK-entry sorted list.
// Candidates tiled through LDS. Scanning ascending j with strict '<' matches
// lax.top_k tie-breaking (lowest index first).
// ---------------------------------------------------------------------------
template <int KK, bool EXCL, bool WANTD>
__global__ __launch_bounds__(256) void knn_kernel(
    const float2* __restrict__ q, int nq,
    const float2* __restrict__ s, int ns,
    int* __restrict__ oidx, float* __restrict__ od2)
{
  __shared__ float2 tile[256];
  int i = blockIdx.x * blockDim.x + threadIdx.x;
  float2 qi = (i < nq) ? q[i] : make_float2(0.f, 0.f);
  float bd[KK];
  int   bi[KK];
#pragma unroll
  for (int r = 0; r < KK; ++r) { bd[r] = 3.0e38f; bi[r] = 0; }

  for (int base = 0; base < ns; base += 256) {
    int j = base + (int)threadIdx.x;
    if (j < ns) tile[threadIdx.x] = s[j];
    __syncthreads();
    int lim = ns - base; if (lim > 256) lim = 256;
    for (int t = 0; t < lim; ++t) {
      int j2 = base + t;
      if (EXCL && j2 == i) continue;
      float dx = qi.x - tile[t].x;
      float dy = qi.y - tile[t].y;
      float d = dx * dx + dy * dy;
      if (d < bd[KK - 1]) {
        float dd = d; int jj = j2;
#pragma unroll
        for (int r = 0; r < KK; ++r) {
          if (dd < bd[r]) {
            float td = bd[r]; int ti = bi[r];
            bd[r] = dd; bi[r] = jj; dd = td; jj = ti;
          }
        }
      }
    }
    __syncthreads();
  }
  if (i < nq) {
#pragma unroll
    for (int r = 0; r < KK; ++r) {
      oidx[i * KK + r] = bi[r];
      if (WANTD) od2[i * KK + r] = bd[r];
    }
  }
}

// ---------------------------------------------------------------------------
// Deterministic farthest point sampling (start at 0), single workgroup.
// mind[] lives in LDS (8192 floats = 32 KB, fine for 320 KB WGP LDS).
// Matches lax.scan semantics: samp[0]=0, samp[t]=argmax after t updates.
// ---------------------------------------------------------------------------
__global__ __launch_bounds__(1024) void fps_kernel(
    const float2* __restrict__ pos, int n, int m, int* __restrict__ samp)
{
  __shared__ float mind[N0];
  __shared__ float shv[1024];
  __shared__ int   shi[1024];
  int tid = (int)threadIdx.x, nt = (int)blockDim.x;
  for (int i = tid; i < n; i += nt) mind[i] = 3.0e38f;
  if (tid == 0) samp[0] = 0;
  __syncthreads();
  int last = 0;
  for (int t = 1; t < m; ++t) {
    float2 pl = pos[last];
    float bv = -1.f; int bidx = 0;
    for (int i = tid; i < n; i += nt) {
      float dx = pos[i].x - pl.x, dy = pos[i].y - pl.y;
      float mv = fminf(mind[i], dx * dx + dy * dy);
      mind[i] = mv;
      if (mv > bv) { bv = mv; bidx = i; }  // ascending i -> first max kept
    }
    shv[tid] = bv; shi[tid] = bidx;
    __syncthreads();
    for (int s = nt >> 1; s > 0; s >>= 1) {
      if (tid < s) {
        if (shv[tid + s] > shv[tid] ||
            (shv[tid + s] == shv[tid] && shi[tid + s] < shi[tid])) {
          shv[tid] = shv[tid + s]; shi[tid] = shi[tid + s];
        }
      }
      __syncthreads();
    }
    last = shi[0];
    if (tid == 0) samp[t] = last;
    __syncthreads();
  }
}

// Edge list: e = i*17+k ; j = nbr[i][k] for k<16, j=i for k=16 (self loop).
__global__ __launch_bounds__(256) void build_edges_kernel(
    const float2* __restrict__ pos, const int* __restrict__ nbr, int n,
    int* __restrict__ ej, float* __restrict__ rel)
{
  int e = blockIdx.x * blockDim.x + threadIdx.x;
  if (e >= n * KE) return;
  int i = e / KE, k = e % KE;
  int j = (k < KNBR) ? nbr[i * KNBR + k] : i;
  ej[e] = j;
  float2 pi = pos[i], pj = pos[j];
  rel[e * 2 + 0] = pi.x - pj.x;
  rel[e * 2 + 1] = pi.y - pj.y;
}

// alpha_in[e,c] = a_i[i,c] - a_src[j,c] + delta[e,c]
__global__ __launch_bounds__(256) void alpha_in_kernel(
    const float* __restrict__ ai, const float* __restrict__ asrc,
    const float* __restrict__ delta, const int* __restrict__ ej,
    int n, int C, float* __restrict__ out)
{
  int t = blockIdx.x * blockDim.x + threadIdx.x;
  if (t >= n * KE * C) return;
  int e = t / C, c = t % C;
  int i = e / KE, j = ej[e];
  out[t] = ai[i * C + c] - asrc[j * C + c] + delta[t];
}

// Per-(node,channel) softmax over 17 incoming edges, then max of w*(v_j+delta).
__global__ __launch_bounds__(256) void attn_aggr_kernel(
    const float* __restrict__ alpha, const float* __restrict__ delta,
    const float* __restrict__ aval, const int* __restrict__ ej,
    int n, int C, float* __restrict__ out)
{
  int t = blockIdx.x * blockDim.x + threadIdx.x;
  if (t >= n * C) return;
  int i = t / C, c = t % C;
  float mx = -3.0e38f;
  for (int k = 0; k < KE; ++k) mx = fmaxf(mx, alpha[(i * KE + k) * C + c]);
  float ssum = 0.f, best = -3.0e38f;
  for (int k = 0; k < KE; ++k) {
    int e = i * KE + k;
    float w = expf(alpha[e * C + c] - mx);
    ssum += w;
    float v = aval[ej[e] * C + c] + delta[e * C + c];
    best = fmaxf(best, w * v);  // max(w_k*v_k)/sum == max((e_k/sum)*v_k)
  }
  out[t] = best / ssum;
}

// TransitionDown pooling: out[s,c] = max_k h[nb[s,k], c]
__global__ __launch_bounds__(256) void gather_max_kernel(
    const float* __restrict__ H, const int* __restrict__ nb,
    int nlo, int C, float* __restrict__ out)
{
  int t = blockIdx.x * blockDim.x + threadIdx.x;
  if (t >= nlo * C) return;
  int s = t / C, c = t % C;
  float m = -3.0e38f;
  for (int k = 0; k < KE; ++k) m = fmaxf(m, H[nb[s * KE + k] * C + c]);
  out[t] = m;
}

__global__ __launch_bounds__(256) void gather_pos_kernel(
    const float2* __restrict__ pos, const int* __restrict__ samp,
    int n, float2* __restrict__ out)
{
  int s = blockIdx.x * blockDim.x + threadIdx.x;
  if (s >= n) return;
  out[s] = pos[samp[s]];
}

// knn_interpolate(k=3): out = skip + sum_k w_k x_sub[idx_k] / sum_k w_k
__global__ __launch_bounds__(256) void interp_add_kernel(
    const float* __restrict__ xsub, const int* __restrict__ idx,
    const float* __restrict__ d2, const float* __restrict__ skip,
    int n, int C, float* __restrict__ out)
{
  int t = blockIdx.x * blockDim.x + threadIdx.x;
  if (t >= n * C) return;
  int i = t / C, c = t % C;
  float ws = 0.f, acc = 0.f;
#pragma unroll
  for (int k = 0; k < KUP; ++k) {
    float w = 1.0f / fmaxf(d2[i * KUP + k], 1e-16f);
    ws += w;
    acc += w * xsub[idx[i * KUP + k] * C + c];
  }
  out[t] = skip[t] + acc / ws;
}

__global__ __launch_bounds__(256) void log_softmax_kernel(
    const float* __restrict__ L, int n, float* __restrict__ out)
{
  int i = blockIdx.x * blockDim.x + threadIdx.x;
  if (i >= n) return;
  float m = -3.0e38f;
#pragma unroll
  for (int c = 0; c < NCLS; ++c) m = fmaxf(m, L[i * NCLS + c]);
  float s = 0.f;
#pragma unroll
  for (int c = 0; c < NCLS; ++c) s += expf(L[i * NCLS + c] - m);
  float ls = m + logf(s);
#pragma unroll
  for (int c = 0; c < NCLS; ++c) out[i * NCLS + c] = L[i * NCLS + c] - ls;
}

// ---------------------------------------------------------------------------
// Host-side parameter views (consumed in reference dict insertion order).
// ---------------------------------------------------------------------------
struct BNP   { const float *g, *b; };
struct MLP2P { const float *W1, *b1; BNP bn1; const float *W2, *b2; BNP bn2; };
struct TBP   { const float *Win, *bin, *Wout, *bout, *Wsrc, *Wdst, *Wval;
               MLP2P pos_nn, attn_nn; };
struct TUP   { const float *Wsub, *bsub; BNP bnsub; const float *W, *b; BNP bn; };

struct Scratch {
  float *x1, *ai, *asrc, *aval, *att;     // node temps (max 8192*64)
  float *EA, *EB, *ED, *REL;              // edge arenas
  int   *EJ;
  float *MU, *RV;
};

static void gemm(hipStream_t st, const float* A, const float* W, const float* b,
                 float* C, int M, int K, int N, int relu)
{
  if (N % 64 == 0) {
    int tiles = ceil_div(M, 16) * (N / 64);
    gemm_kernel<4><<<ceil_div(tiles, 8), 256, 0, st>>>(A, W, b, C, M, K, N, relu);
  } else {
    int tiles = ceil_div(M, 16) * ceil_div(N, 16);
    gemm_kernel<1><<<ceil_div(tiles, 8), 256, 0, st>>>(A, W, b, C, M, K, N, relu);
  }
}

static void norm(hipStream_t st, float* H, int M, int C, const float* g,
                 const float* b, int relu, float* mu, float* rv)
{
  colstats_kernel<<<C, 256, 0, st>>>(H, M, C, mu, rv);
  int total = M * C;
  bnact_kernel<<<ceil_div(total, 256), 256, 0, st>>>(H, total, C, mu, rv, g, b, relu);
}

// One PointTransformer block, in-place on x[n,c].
static void run_tb(hipStream_t st, const TBP& p, float* x, const float2* pos,
                   const int* nbr, int n, int c, Scratch& S)
{
  int E = n * KE;
  build_edges_kernel<<<ceil_div(E, 256), 256, 0, st>>>(pos, nbr, n, S.EJ, S.REL);
  gemm(st, x, p.Win, p.bin, S.x1, n, c, c, 1);
  gemm(st, S.x1, p.Wdst, nullptr, S.ai,   n, c, c, 0);
  gemm(st, S.x1, p.Wsrc, nullptr, S.asrc, n, c, c, 0);
  gemm(st, S.x1, p.Wval, nullptr, S.aval, n, c, c, 0);
  // pos_nn: 2 -> HID -> c, BN+ReLU after each
  gemm(st, S.REL, p.pos_nn.W1, p.pos_nn.b1, S.EB, E, 2, HIDN, 0);
  norm(st, S.EB, E, HIDN, p.pos_nn.bn1.g, p.pos_nn.bn1.b, 1, S.MU, S.RV);
  gemm(st, S.EB, p.pos_nn.W2, p.pos_nn.b2, S.ED, E, HIDN, c, 0);
  norm(st, S.ED, E, c, p.pos_nn.bn2.g, p.pos_nn.bn2.b, 1, S.MU, S.RV);
  // attn_nn on (a_i - a_j + delta): c -> HID -> c
  alpha_in_kernel<<<ceil_div(E * c, 256), 256, 0, st>>>(S.ai, S.asrc, S.ED, S.EJ, n, c, S.EA);
  gemm(st, S.EA, p.attn_nn.W1, p.attn_nn.b1, S.EB, E, c, HIDN, 0);
  norm(st, S.EB, E, HIDN, p.attn_nn.bn1.g, p.attn_nn.bn1.b, 1, S.MU, S.RV);
  gemm(st, S.EB, p.attn_nn.W2, p.attn_nn.b2, S.EA, E, HIDN, c, 0);
  norm(st, S.EA, E, c, p.attn_nn.bn2.g, p.attn_nn.bn2.b, 1, S.MU, S.RV);
  attn_aggr_kernel<<<ceil_div(n * c, 256), 256, 0, st>>>(S.EA, S.ED, S.aval, S.EJ, n, c, S.att);
  gemm(st, S.att, p.Wout, p.bout, x, n, c, c, 1);
}

extern "C" void kernel_launch(void* const* d_in, const int* in_sizes, int n_in,
                              void* d_out, int out_size, void* d_ws, size_t ws_size,
                              hipStream_t stream)
{
  (void)in_sizes; (void)n_in; (void)out_size; (void)ws_size;
  // ---- parse inputs: x, pos, then params in dict insertion order ----
  int ii = 0;
  auto nx = [&]() -> const float* { return (const float*)d_in[ii++]; };
  const float*  Xin  = nx();
  const float2* Pos0 = (const float2*)d_in[ii++];

  auto rdBN = [&](BNP& o)   { o.g = nx(); o.b = nx(); };
  auto rdM  = [&](MLP2P& o) { o.W1 = nx(); o.b1 = nx(); rdBN(o.bn1);
                              o.W2 = nx(); o.b2 = nx(); rdBN(o.bn2); };
  auto rdTB = [&](TBP& o)   { o.Win = nx(); o.bin = nx(); o.Wout = nx(); o.bout = nx();
                              o.Wsrc = nx(); o.Wdst = nx(); o.Wval = nx();
                              rdM(o.pos_nn); rdM(o.attn_nn); };
  auto rdTU = [&](TUP& o)   { o.Wsub = nx(); o.bsub = nx(); rdBN(o.bnsub);
                              o.W = nx(); o.b = nx(); rdBN(o.bn); };

  const float* miW = nx(); const float* mib = nx(); BNP mibn; rdBN(mibn);
  TBP tb_input;  rdTB(tb_input);
  const float* td0W = nx(); const float* td0b = nx();
  const float* td1W = nx(); const float* td1b = nx();
  TBP tb_down0, tb_down1; rdTB(tb_down0); rdTB(tb_down1);
  const float* sumW = nx(); const float* sumb = nx();
  TBP tb_summit; rdTB(tb_summit);
  TUP tu0, tu1; rdTU(tu0); rdTU(tu1);
  TBP tb_up0, tb_up1; rdTB(tb_up0); rdTB(tb_up1);
  const float* oW1 = nx(); const float* ob1 = nx();
  const float* oW2 = nx(); const float* ob2 = nx();

  // ---- carve workspace ----
  char* w = (char*)d_ws;
  auto alloc_f = [&](size_t cnt) -> float* {
    float* p = (float*)w; w += ((cnt * 4 + 255) / 256) * 256; return p; };
  auto alloc_i = [&](size_t cnt) -> int* {
    int* p = (int*)w; w += ((cnt * 4 + 255) / 256) * 256; return p; };

  Scratch S;
  S.EA  = alloc_f((size_t)E0 * 64);   // >= E1*C1 and E2*C2
  S.EB  = alloc_f((size_t)E0 * 64);
  S.ED  = alloc_f((size_t)E0 * 64);
  S.REL = alloc_f((size_t)E0 * 2);
  S.EJ  = alloc_i((size_t)E0);
  S.x1   = alloc_f((size_t)N0 * C0);
  S.ai   = alloc_f((size_t)N0 * C0);
  S.asrc = alloc_f((size_t)N0 * C0);
  S.aval = alloc_f((size_t)N0 * C0);
  S.att  = alloc_f((size_t)N0 * C0);
  S.MU = alloc_f(256); S.RV = alloc_f(256);

  float* X0   = alloc_f((size_t)N0 * C0);   // skip level 0
  float* X1   = alloc_f((size_t)N1 * C1);   // skip level 1
  float* X2   = alloc_f((size_t)N2 * C2);
  float* XA   = alloc_f((size_t)N0 * C0);   // current-x buffers
  float* XB   = alloc_f((size_t)N0 * C0);
  float* XSUB = alloc_f((size_t)N1 * C1);
  float* XSK  = alloc_f((size_t)N0 * C0);
  float* HT   = alloc_f((size_t)N0 * C1);   // TransitionDown pre-pool features
  float* HH   = alloc_f((size_t)N0 * HIDN);
  float* LG   = alloc_f((size_t)N0 * NCLS);
  float2* P1  = (float2*)alloc_f((size_t)N1 * 2);
  float2* P2  = (float2*)alloc_f((size_t)N2 * 2);
  float* D2I  = alloc_f((size_t)N0 * KUP);
  int* NB0   = alloc_i((size_t)N0 * KNBR);
  int* NB1   = alloc_i((size_t)N1 * KNBR);
  int* NB2   = alloc_i((size_t)N2 * KNBR);
  int* NBTD  = alloc_i((size_t)N1 * KE);
  int* SAMP1 = alloc_i((size_t)N1);
  int* SAMP2 = alloc_i((size_t)N2);
  int* IIDX  = alloc_i((size_t)N0 * KUP);

  // ---- mlp_input: relu(BN(x @ W + b)) ----
  gemm(stream, Xin, miW, mib, X0, N0, 3, C0, 0);
  norm(stream, X0, N0, C0, mibn.g, mibn.b, 1, S.MU, S.RV);

  // ---- level-0 graph + input transformer block ----
  knn_kernel<KNBR, true, false><<<ceil_div(N0, 256), 256, 0, stream>>>(Pos0, N0, Pos0, N0, NB0, nullptr);
  run_tb(stream, tb_input, X0, Pos0, NB0, N0, C0, S);

  // ---- TransitionDown 0: 8192 -> 2048, C 64 -> 128 ----
  fps_kernel<<<1, 1024, 0, stream>>>(Pos0, N0, N1, SAMP1);
  gather_pos_kernel<<<ceil_div(N1, 256), 256, 0, stream>>>(Pos0, SAMP1, N1, P1);
  knn_kernel<KE, false, false><<<ceil_div(N1, 256), 256, 0, stream>>>(P1, N1, Pos0, N0, NBTD, nullptr);
  gemm(stream, X0, td0W, td0b, HT, N0, C0, C1, 0);
  norm(stream, HT, N0, C1, nullptr, nullptr, 1, S.MU, S.RV);   // instance norm
  gather_max_kernel<<<ceil_div(N1 * C1, 256), 256, 0, stream>>>(HT, NBTD, N1, C1, X1);
  knn_kernel<KNBR, true, false><<<ceil_div(N1, 256), 256, 0, stream>>>(P1, N1, P1, N1, NB1, nullptr);
  run_tb(stream, tb_down0, X1, P1, NB1, N1, C1, S);

  // ---- TransitionDown 1: 2048 -> 512, C 128 -> 256 ----
  fps_kernel<<<1, 1024, 0, stream>>>(P1, N1, N2, SAMP2);
  gather_pos_kernel<<<ceil_div(N2, 256), 256, 0, stream>>>(P1, SAMP2, N2, P2);
  knn_kernel<KE, false, false><<<ceil_div(N2, 256), 256, 0, stream>>>(P2, N2, P1, N1, NBTD, nullptr);
  gemm(stream, X1, td1W, td1b, HT, N1, C1, C2, 0);
  norm(stream, HT, N1, C2, nullptr, nullptr, 1, S.MU, S.RV);
  gather_max_kernel<<<ceil_div(N2 * C2, 256), 256, 0, stream>>>(HT, NBTD, N2, C2, X2);
  knn_kernel<KNBR, true, false><<<ceil_div(N2, 256), 256, 0, stream>>>(P2, N2, P2, N2, NB2, nullptr);
  run_tb(stream, tb_down1, X2, P2, NB2, N2, C2, S);

  // ---- summit ----
  gemm(stream, X2, sumW, sumb, XA, N2, C2, C2, 1);
  run_tb(stream, tb_summit, XA, P2, NB2, N2, C2, S);

  // ---- decoder step 0: level2 -> level1 (tu[1], tb_up[1]) ----
  gemm(stream, XA, tu1.Wsub, tu1.bsub, XSUB, N2, C2, C1, 0);
  norm(stream, XSUB, N2, C1, tu1.bnsub.g, tu1.bnsub.b, 1, S.MU, S.RV);
  knn_kernel<KUP, false, true><<<ceil_div(N1, 256), 256, 0, stream>>>(P1, N1, P2, N2, IIDX, D2I);
  gemm(stream, X1, tu1.W, tu1.b, XSK, N1, C1, C1, 0);
  norm(stream, XSK, N1, C1, tu1.bn.g, tu1.bn.b, 1, S.MU, S.RV);
  interp_add_kernel<<<ceil_div(N1 * C1, 256), 256, 0, stream>>>(XSUB, IIDX, D2I, XSK, N1, C1, XB);
  run_tb(stream, tb_up1, XB, P1, NB1, N1, C1, S);

  // ---- decoder step 1: level1 -> level0 (tu[0], tb_up[0]) ----
  gemm(stream, XB, tu0.Wsub, tu0.bsub, XSUB, N1, C1, C0, 0);
  norm(stream, XSUB, N1, C0, tu0.bnsub.g, tu0.bnsub.b, 1, S.MU, S.RV);
  knn_kernel<KUP, false, true><<<ceil_div(N0, 256), 256, 0, stream>>>(Pos0, N0, P1, N1, IIDX, D2I);
  gemm(stream, X0, tu0.W, tu0.b, XSK, N0, C0, C0, 0);
  norm(stream, XSK, N0, C0, tu0.bn.g, tu0.bn.b, 1, S.MU, S.RV);
  interp_add_kernel<<<ceil_div(N0 * C0, 256), 256, 0, stream>>>(XSUB, IIDX, D2I, XSK, N0, C0, XA);
  run_tb(stream, tb_up0, XA, Pos0, NB0, N0, C0, S);

  // ---- head + log_softmax ----
  gemm(stream, XA, oW1, ob1, HH, N0, C0, HIDN, 1);
  gemm(stream, HH, oW2, ob2, LG, N0, HIDN, NCLS, 0);
  log_softmax_kernel<<<ceil_div(N0, 256), 256, 0, stream>>>(LG, N0, (float*)d_out);
}